// Crf_44607530336616
// MI455X (gfx1250) — compile-verified
//
#include <hip/hip_runtime.h>

// CRF loss on gfx1250 (MI455X, wave32), two kernels.
//
// Kernel 1 (crf_score): label-path score, 1 wave per batch, coalesced label
//   loads, wave-shuffle reduction. Writes out[b] = -score.
//
// Kernel 2 (crf_fwd): forward-algorithm scan, 1 wave per 16-batch tile.
//   Scaled (multiply/renormalize) forward recurrence:
//      alpha_t = M + log(u_t),  u_t = (E . p_{t-1}) * exp(x_t),  E = exp(trans)
//   with renormalization (u/umax, M += log umax) only every 8 steps.
//   The 16x16x16 f32 matmul per step = 4x V_WMMA_F32_16X16X4_F32 in two
//   independent accumulator chains.
//
//   Label axis is stored permuted (sigma = reg r, half h -> 4*(r>>1)+(r&1)+2h)
//   so the WMMA C/D output layout IS the next step's B-operand layout:
//   B_k = {p[2k], p[2k+1]} with zero cross-lane traffic. exp(x_t) is
//   precomputed one iteration ahead (off the serial chain).

typedef __attribute__((ext_vector_type(2))) float v2f;
typedef __attribute__((ext_vector_type(8))) float v8f;

#define TT 512
#define LL 16

// SWAPX16: ds_swizzle group-of-32, xor_mask=0x10, and_mask=0x1f
__device__ __forceinline__ float half_swap_f(float v) {
    return __int_as_float(__builtin_amdgcn_ds_swizzle(__float_as_int(v), 0x401F));
}

__device__ __forceinline__ void load_xfrag(const float* base2h, int t, float xf[8]) {
    const float* p = base2h + (size_t)t * LL;
#pragma unroll
    for (int i = 0; i < 4; ++i) {
        float2 v = *(const float2*)(p + 4 * i);   // labels 4i+2h, 4i+2h+1
        xf[2 * i]     = v.x;
        xf[2 * i + 1] = v.y;
    }
}

// ---------------- label-path score: out[b] = -(point + trans) ----------------
__launch_bounds__(256)
__global__ void crf_score_kernel(const float* __restrict__ x,
                                 const float* __restrict__ trans,
                                 const int*   __restrict__ label,
                                 const int*   __restrict__ length,
                                 float*       __restrict__ out)
{
    const int lane = threadIdx.x & 31;
    const int b    = blockIdx.x * 8 + (threadIdx.x >> 5);
    const int len  = length[b];
    const size_t base = (size_t)b * TT;

    float acc = 0.0f;
    for (int t = lane; t < TT; t += 32) {
        if (t < len) {
            const int lab = label[base + t];
            acc += x[(base + t) * LL + lab];
            if (t > 0) acc += trans[label[base + t - 1] * LL + lab];
        }
    }
#pragma unroll
    for (int off = 16; off; off >>= 1) acc += __shfl_xor(acc, off, 32);
    if (lane == 0) out[b] = -acc;
}

// ---------------- forward scan: out[b] += logsumexp(alpha[len-1]) ------------
__launch_bounds__(32)
__global__ void crf_fwd_kernel(const float* __restrict__ x,
                               const float* __restrict__ trans,
                               const int*   __restrict__ length,
                               float*       __restrict__ out)
{
    const int lane = threadIdx.x & 31;
    const int half = lane >> 4;
    const int m    = lane & 15;          // batch-in-tile (N column)
    const int bm   = blockIdx.x * 16 + m;

    // output-label permutation: M-slot s -> label jo(s)
    const int s  = lane & 15;
    const int jo = 4 * ((s >> 1) & 3) + (s & 1) + 2 * (s >> 3);

    // constant A operand: A'[M][kappa] = E[kappa][jo(M)] = exp(trans[kappa][jo(M)])
    // fragment k: v0 -> kappa = 4k + 2*half, v1 -> kappa = 4k+1 + 2*half
    v2f A[4];
#pragma unroll
    for (int k = 0; k < 4; ++k) {
        const int k0 = 4 * k + 2 * half;
        A[k].x = __expf(trans[k0 * LL + jo]);
        A[k].y = __expf(trans[(k0 + 1) * LL + jo]);
    }

    const int len = length[bm];
    const float* xb = x + (size_t)bm * TT * LL + 2 * half;  // permuted frag base

    // ---- t = 0 : alpha0 = x0  ->  u0 = exp(x0), M = 0 ----
    float p[8], finu[8];
    {
        float xf0[8];
        load_xfrag(xb, 0, xf0);
#pragma unroll
        for (int r = 0; r < 8; ++r) { p[r] = __expf(xf0[r]); finu[r] = p[r]; }
    }
    float M = 0.0f, finm = 0.0f;

    // pipeline: exc = exp(x_t) for the iteration about to run
    float exc[8];
    {
        float xf1[8];
        load_xfrag(xb, 1, xf1);
#pragma unroll
        for (int r = 0; r < 8; ++r) exc[r] = __expf(xf1[r]);
    }

    const v8f z8 = {};
    for (int t = 1; t < TT; ++t) {
        // issue next x loads early (consumed at end of iteration)
        float xn[8];
        if (t + 1 < TT) load_xfrag(xb, t + 1, xn);
        if (t + 16 < TT) __builtin_prefetch(xb + (size_t)(t + 16) * LL, 0, 1);

        // S = E^T x p : two independent WMMA chains
        v2f B0 = {p[0], p[1]}, B1 = {p[2], p[3]};
        v2f B2 = {p[4], p[5]}, B3 = {p[6], p[7]};
        v8f accA = __builtin_amdgcn_wmma_f32_16x16x4_f32(false, A[0], false, B0,
                                                         (short)0, z8, false, false);
        v8f accB = __builtin_amdgcn_wmma_f32_16x16x4_f32(false, A[1], false, B1,
                                                         (short)0, z8, false, false);
        accA = __builtin_amdgcn_wmma_f32_16x16x4_f32(false, A[2], false, B2,
                                                     (short)0, accA, false, false);
        accB = __builtin_amdgcn_wmma_f32_16x16x4_f32(false, A[3], false, B3,
                                                     (short)0, accB, false, false);

        // u_t = S * exp(x_t)    (alpha_t = M + log u_t)
        float u[8];
#pragma unroll
        for (int r = 0; r < 8; ++r) u[r] = (accA[r] + accB[r]) * exc[r];

        // capture at t == len-1 (before any renormalization of u_t)
        const bool cap = (t == len - 1);
        finm = cap ? M : finm;
#pragma unroll
        for (int r = 0; r < 8; ++r) finu[r] = cap ? u[r] : finu[r];

        // renormalize every 8 steps (E>=1, exp(x)<=e^6 keeps f32 in range)
        if ((t & 7) == 0) {
            float um = fmaxf(u[0], u[1]);
#pragma unroll
            for (int r = 2; r < 8; ++r) um = fmaxf(um, u[r]);
            um = fmaxf(um, half_swap_f(um));
            const float ir = __builtin_amdgcn_rcpf(um);
#pragma unroll
            for (int r = 0; r < 8; ++r) p[r] = u[r] * ir;
            M += __logf(um);
        } else {
#pragma unroll
            for (int r = 0; r < 8; ++r) p[r] = u[r];
        }

        // exp for next iteration (off the serial chain)
        if (t + 1 < TT) {
#pragma unroll
            for (int r = 0; r < 8; ++r) exc[r] = __expf(xn[r]);
        }
    }

    // z = finm + log(sum_j finu[j])
    float ssum = finu[0] + finu[1];
#pragma unroll
    for (int r = 2; r < 8; ++r) ssum += finu[r];
    ssum += half_swap_f(ssum);
    const float zv = finm + __logf(ssum);

    if (lane < 16) out[bm] = zv + out[bm];   // out already holds -score
}

extern "C" void kernel_launch(void* const* d_in, const int* in_sizes, int n_in,
                              void* d_out, int out_size, void* d_ws, size_t ws_size,
                              hipStream_t stream) {
    const float* x      = (const float*)d_in[0];   // (B,T,L) f32
    const float* trans  = (const float*)d_in[1];   // (L,L)   f32
    const int*   label  = (const int*)d_in[2];     // (B,T)   i32
    const int*   length = (const int*)d_in[3];     // (B,)    i32
    float*       out    = (float*)d_out;           // (B,1)   f32

    const int B = in_sizes[3];                     // 4096
    crf_score_kernel<<<B / 8, 256, 0, stream>>>(x, trans, label, length, out);
    crf_fwd_kernel<<<B / 16, 32, 0, stream>>>(x, trans, length, out);
}